// loss_whole_6846177870488
// MI455X (gfx1250) — compile-verified
//
#include <hip/hip_runtime.h>
#include <math.h>

// ---- problem dims (fixed by reference setup_inputs) ----
#define B_ 32
#define N_ 4096
#define K_ 16
#define D_ 128
#define PTS_PER_BLOCK 128

typedef float v2f __attribute__((ext_vector_type(2)));
typedef float v8f __attribute__((ext_vector_type(8)));

// sign(v)*|v|^m via fast exp/log (throughput: v_exp_f32/v_log_f32)
__device__ __forceinline__ float sgnpow(float v, float m) {
    float a = fabsf(v);
    float p = (a > 0.0f) ? __expf(m * __logf(a)) : 0.0f;
    return copysignf(p, v);
}

__global__ void init_kernel(float* ws) { ws[0] = 0.0f; }

// ---------------- assign_matrix.sum over N -> S[b][k] ----------------
__global__ __launch_bounds__(256)
void asum_kernel(const float* __restrict__ assign, float* __restrict__ S) {
    __shared__ float tmp[256];
    const int b = blockIdx.x;
    const int t = threadIdx.x;
    const int k  = t & 15;
    const int nb = t >> 4;
    const float* base = assign + (size_t)b * N_ * K_;
    float acc = 0.0f;
    // addr = base + t + 256*j : fully coalesced across the block
    for (int n = nb; n < N_; n += 16) acc += base[(size_t)n * K_ + k];
    tmp[t] = acc;
    __syncthreads();
    if (t < 16) {
        float s = 0.0f;
        for (int j = 0; j < 16; ++j) s += tmp[j * 16 + t];
        S[b * K_ + t] = s;
    }
}

// ---------------- heavy REC kernel ----------------
// grid = (N/128, B), block = 128 (4 waves). Each wave owns 32 points,
// processed as two 16-point WMMA groups. 9 x v_wmma_f32_16x16x4_f32 per
// group compute R^T p for psi/pci/ni across all 16 superquadrics.
__global__ __launch_bounds__(128)
void rec_kernel(const float* __restrict__ pc,
                const float* __restrict__ normals,
                const float* __restrict__ randn_dis,
                const float* __restrict__ scale,
                const float* __restrict__ shapes,
                const float* __restrict__ rotate,
                const float* __restrict__ pc_mean,
                const float* __restrict__ assign,
                float* __restrict__ rec_acc) {
    __shared__ float s_rot[K_][9];     // rotate[b][k][j][i], row-major j*3+i
    __shared__ float s_rtm[K_][3];     // R^T * pc_assign_mean
    __shared__ float s_scale[K_][3];
    __shared__ float s_shape[K_][2];
    __shared__ float psiS[4][16][49];  // stride 49 -> conflict-free banks
    __shared__ float pciS[4][16][49];
    __shared__ float niS [4][16][49];
    __shared__ float red[128];

    const int b    = blockIdx.y;
    const int n0   = blockIdx.x * PTS_PER_BLOCK;
    const int tid  = threadIdx.x;
    const int wave = tid >> 5;
    const int lane = tid & 31;
    const bool lo  = lane < 16;

    // stage per-b constants
    if (tid < K_ * 9) s_rot[tid / 9][tid % 9]   = rotate[(size_t)b * K_ * 9 + tid];
    if (tid < K_ * 3) s_scale[tid / 3][tid % 3] = scale[(size_t)b * K_ * 3 + tid];
    if (tid < K_ * 2) s_shape[tid / 2][tid % 2] = shapes[(size_t)b * K_ * 2 + tid];
    __syncthreads();
    if (tid < K_ * 3) {
        int k = tid / 3, i = tid % 3;
        float acc = 0.0f;
        for (int j = 0; j < 3; ++j) acc += s_rot[k][j * 3 + i] * pc_mean[((size_t)b * K_ + k) * 3 + j];
        s_rtm[k][i] = acc;
    }
    __syncthreads();

    // B matrices: 4x16, column c = t*16 + (lane&15) = flat (k*3+i), rows j=0..3
    // (row 3 zero pad). psi_i = sum_j R[j][i] * v_j  =>  B[j][c] = R[j][i].
    // Layout assumption (mirrors documented 16x4 A layout transposed):
    //   vgpr0 = row (lo?0:1), vgpr1 = row (lo?2:3)
    v2f Bm[3];
    {
        int col = lane & 15;
        for (int t = 0; t < 3; ++t) {
            int c = t * 16 + col;
            int k = c / 3, i = c - 3 * k;
            v2f bb;
            bb.x = s_rot[k][(lo ? 0 : 1) * 3 + i];
            bb.y = lo ? s_rot[k][2 * 3 + i] : 0.0f;
            Bm[t] = bb;
        }
    }

    // own point
    const int n_own = n0 + wave * 32 + lane;
    const float* pcp = pc      + (size_t)(b * N_ + n_own) * 3;
    const float* nrp = normals + (size_t)(b * N_ + n_own) * 3;
    float px = pcp[0], py = pcp[1], pz = pcp[2];
    float nx = nrp[0], ny = nrp[1], nz = nrp[2];
    float rd = randn_dis[(size_t)b * N_ + n_own] * 0.01f;           // W_STD
    float sx = px + rd * nx, sy = py + rd * ny, sz = pz + rd * nz;  // pc_sample
    float inl = rsqrtf(nx * nx + ny * ny + nz * nz);
    float ux = nx * inl, uy = ny * inl, uz = nz * inl;              // unit normal

    // prefetch assign rows for this wave's two groups (global_prefetch_b8)
    __builtin_prefetch(assign + ((size_t)(b * N_ + n0 + wave * 32 + (lane & 15)) * K_), 0, 3);
    __builtin_prefetch(assign + ((size_t)(b * N_ + n0 + wave * 32 + 16 + (lane & 15)) * K_), 0, 3);

    float acc = 0.0f;
    for (int g = 0; g < 2; ++g) {
        // A (16x4): lanes 0-15 hold (x,y) of point M=lane; lanes 16-31 hold (z,0) of M=lane-16
        int srcl = g * 16 + (lane & 15);
        float aPx = __shfl(px, srcl, 32), aPy = __shfl(py, srcl, 32), aPz = __shfl(pz, srcl, 32);
        float aSx = __shfl(sx, srcl, 32), aSy = __shfl(sy, srcl, 32), aSz = __shfl(sz, srcl, 32);
        float aUx = __shfl(ux, srcl, 32), aUy = __shfl(uy, srcl, 32), aUz = __shfl(uz, srcl, 32);
        v2f Apc, Asm, Anm;
        Apc.x = lo ? aPx : aPz;  Apc.y = lo ? aPy : 0.0f;
        Asm.x = lo ? aSx : aSz;  Asm.y = lo ? aSy : 0.0f;
        Anm.x = lo ? aUx : aUz;  Anm.y = lo ? aUy : 0.0f;

        for (int t = 0; t < 3; ++t) {
            v8f z = 0.0f;
            v8f dps = __builtin_amdgcn_wmma_f32_16x16x4_f32(false, Asm, false, Bm[t], (short)0, z, false, false);
            v8f dpc = __builtin_amdgcn_wmma_f32_16x16x4_f32(false, Apc, false, Bm[t], (short)0, z, false, false);
            v8f dni = __builtin_amdgcn_wmma_f32_16x16x4_f32(false, Anm, false, Bm[t], (short)0, z, false, false);
            // D layout: lanes 0-15: M=r, N=lane; lanes 16-31: M=r+8, N=lane-16
            int colf = t * 16 + (lane & 15);
            int rb = lo ? 0 : 8;
            for (int r = 0; r < 8; ++r) {
                psiS[wave][rb + r][colf] = dps[r];
                pciS[wave][rb + r][colf] = dpc[r];
                niS [wave][rb + r][colf] = dni[r];
            }
        }
        __syncthreads();

        // 2 lanes per point: lane&15 -> point, lane/16 -> k half
        const int m = lane & 15;
        const int n = n0 + wave * 32 + g * 16 + m;
        const int kstart = lo ? 0 : 8;
        const float* arow = assign + ((size_t)(b * N_ + n) * K_ + kstart);
        for (int kk = 0; kk < 8; ++kk) {
            const int k = kstart + kk;
            float p0 = psiS[wave][m][3 * k + 0] - s_rtm[k][0];
            float p1 = psiS[wave][m][3 * k + 1] - s_rtm[k][1];
            float p2 = psiS[wave][m][3 * k + 2] - s_rtm[k][2];
            float q0 = pciS[wave][m][3 * k + 0] - s_rtm[k][0];
            float q1 = pciS[wave][m][3 * k + 1] - s_rtm[k][1];
            float q2 = pciS[wave][m][3 * k + 2] - s_rtm[k][2];
            float w0 = niS[wave][m][3 * k + 0];
            float w1 = niS[wave][m][3 * k + 1];
            float w2 = niS[wave][m][3 * k + 2];

            // argmax over (-w0, w0, -w1, w1, -w2, w2) -> face axis+sign (first max)
            float best = -w0; int axis = 0; float sgn = -1.0f;
            if ( w0 > best) { best =  w0; axis = 0; sgn =  1.0f; }
            if (-w1 > best) { best = -w1; axis = 1; sgn = -1.0f; }
            if ( w1 > best) { best =  w1; axis = 1; sgn =  1.0f; }
            if (-w2 > best) { best = -w2; axis = 2; sgn = -1.0f; }
            if ( w2 > best) { best =  w2; axis = 2; sgn =  1.0f; }

            float s0 = s_scale[k][0], s1 = s_scale[k][1], s2 = s_scale[k][2];
            float pp0 = (axis == 0) ? sgn * s0 : p0;
            float pp1 = (axis == 1) ? sgn * s1 : p1;
            float pp2 = (axis == 2) ? sgn * s2 : p2;
            pp0 = fminf(fmaxf(pp0, -s0), s0);
            pp1 = fminf(fmaxf(pp1, -s1), s1);
            pp2 = fminf(fmaxf(pp2, -s2), s2);

            // sin/cos(theta), sin/cos(phi) without atan2/sin/cos:
            // theta = atan2(pp1, pp0); phi = atan2(pp2, rxy), rxy >= 0
            float rxy2 = pp0 * pp0 + pp1 * pp1;
            float rxy  = sqrtf(rxy2);
            float ct, st;
            if (rxy > 0.0f) { float iv = 1.0f / rxy; ct = pp0 * iv; st = pp1 * iv; }
            else            { ct = 1.0f; st = 0.0f; }   // rxy==0 => pp0=pp1=0 => theta=0
            float rr = sqrtf(rxy2 + pp2 * pp2);
            float cp, sp;
            if (rr > 0.0f)  { float iv = 1.0f / rr; cp = rxy * iv; sp = pp2 * iv; }
            else            { cp = 1.0f; sp = 0.0f; }   // r==0 => phi=0

            float e0 = s_shape[k][0], e1 = s_shape[k][1];
            float gph = sgnpow(cp, e0);
            float fph = sgnpow(sp, e0);
            float gth = sgnpow(ct, e1);
            float fth = sgnpow(st, e1);
            float X = s0 * gph * gth;
            float Y = s1 * gph * fth;
            float Z = s2 * fph;
            float d0 = X - q0, d1 = Y - q1, d2 = Z - q2;
            acc += (d0 * d0 + d1 * d1 + d2 * d2) * arow[kk];
        }
        __syncthreads();
    }

    // block reduce -> atomic into ws[0]
    red[tid] = acc;
    __syncthreads();
    for (int off = 64; off > 0; off >>= 1) {
        if (tid < off) red[tid] += red[tid + off];
        __syncthreads();
    }
    if (tid == 0) atomicAdd(rec_acc, red[0]);
}

// ---------------- finalize: combine all terms ----------------
__device__ __forceinline__ float block_reduce256(float v, float* s) {
    const int t = threadIdx.x;
    s[t] = v;
    __syncthreads();
    for (int off = 128; off > 0; off >>= 1) {
        if (t < off) s[t] += s[t + off];
        __syncthreads();
    }
    float r = s[0];
    __syncthreads();
    return r;
}

__global__ __launch_bounds__(256)
void finalize_kernel(const float* __restrict__ exist,
                     const float* __restrict__ mu,
                     const float* __restrict__ log_var,
                     const float* __restrict__ pc_mean,
                     const float* __restrict__ trans,
                     const float* __restrict__ ws,
                     float* __restrict__ out) {
    __shared__ float sacc[256];
    const int t = threadIdx.x;
    const float* S = ws + 16;  // assign column sums, B*K

    // KLD: sum over (b,d) of (1 + lv - mu^2 - e^lv)
    float kld = 0.0f;
    for (int i = t; i < B_ * D_; i += 256) {
        float lv = log_var[i], m = mu[i];
        kld += 1.0f + lv - m * m - __expf(lv);
    }
    kld = block_reduce256(kld, sacc);

    // EXT (BCE-with-logits vs gt = S>24) and CST (||mean - trans||)
    float ext = 0.0f, cst = 0.0f;
    for (int i = t; i < B_ * K_; i += 256) {
        float l = exist[i];
        float gt = (S[i] > 24.0f) ? 1.0f : 0.0f;
        ext += fmaxf(l, 0.0f) - l * gt + log1pf(__expf(-fabsf(l)));
        float d0 = pc_mean[i * 3 + 0] - trans[i * 3 + 0];
        float d1 = pc_mean[i * 3 + 1] - trans[i * 3 + 1];
        float d2 = pc_mean[i * 3 + 2] - trans[i * 3 + 2];
        cst += sqrtf(d0 * d0 + d1 * d1 + d2 * d2);
    }
    ext = block_reduce256(ext, sacc);
    cst = block_reduce256(cst, sacc);

    // SPS: mean_b (mean_k sqrt(S/N + 0.01))^2
    float sps = 0.0f;
    if (t < B_) {
        float m = 0.0f;
        for (int k = 0; k < K_; ++k) m += sqrtf(S[t * K_ + k] * (1.0f / N_) + 0.01f);
        m *= (1.0f / K_);
        sps = m * m;
    }
    sps = block_reduce256(sps, sacc);

    if (t == 0) {
        float REC = ws[0] * (1.0f / ((float)B_ * (float)N_));
        float SPS = sps * (1.0f / B_);
        float EXT = ext * (1.0f / (B_ * K_));
        float KLD = (-0.5f * kld) * (1.0f / B_);
        float CST = cst * (1.0f / (B_ * K_));
        out[0] = 1.0f * REC + 0.1f * SPS + 0.01f * EXT + 0.001f * KLD + 0.1f * CST;
    }
}

extern "C" void kernel_launch(void* const* d_in, const int* in_sizes, int n_in,
                              void* d_out, int out_size, void* d_ws, size_t ws_size,
                              hipStream_t stream) {
    const float* pc        = (const float*)d_in[0];
    const float* normals   = (const float*)d_in[1];
    const float* randn_dis = (const float*)d_in[2];
    const float* scale     = (const float*)d_in[3];
    const float* shapes    = (const float*)d_in[4];
    const float* rotate    = (const float*)d_in[5];
    const float* pc_mean   = (const float*)d_in[6];
    const float* assign    = (const float*)d_in[7];
    const float* exist     = (const float*)d_in[8];
    const float* mu        = (const float*)d_in[9];
    const float* log_var   = (const float*)d_in[10];
    const float* trans     = (const float*)d_in[11];
    float* ws  = (float*)d_ws;     // [0]=REC acc, [16..16+512)=assign col sums
    float* out = (float*)d_out;

    init_kernel<<<1, 1, 0, stream>>>(ws);
    asum_kernel<<<B_, 256, 0, stream>>>(assign, ws + 16);
    rec_kernel<<<dim3(N_ / PTS_PER_BLOCK, B_), PTS_PER_BLOCK, 0, stream>>>(
        pc, normals, randn_dis, scale, shapes, rotate, pc_mean, assign, ws);
    finalize_kernel<<<1, 256, 0, stream>>>(exist, mu, log_var, pc_mean, trans, ws, out);
}